// SimpleRecGNN_13898514170321
// MI455X (gfx1250) — compile-verified
//
#include <hip/hip_runtime.h>

typedef float v2f __attribute__((ext_vector_type(2)));
typedef float v8f __attribute__((ext_vector_type(8)));

// ---------------------------------------------------------------------------
// Preprocessing: int64 edges -> int32, degree count, dinv, per-edge norm
// ---------------------------------------------------------------------------
__global__ void k_init_deg(float* __restrict__ deg, int N) {
  int i = blockIdx.x * blockDim.x + threadIdx.x;
  if (i < N) deg[i] = 1.0f;  // self-loop contributes 1 to in-degree
}

__global__ void k_edge_prep(const long long* __restrict__ ei,
                            int* __restrict__ src32, int* __restrict__ dst32,
                            float* __restrict__ deg, int E) {
  int e = blockIdx.x * blockDim.x + threadIdx.x;
  if (e < E) {
    int s = (int)ei[e];                 // row 0: src
    int d = (int)ei[(long long)E + e];  // row 1: dst
    src32[e] = s;
    dst32[e] = d;
    unsafeAtomicAdd(&deg[d], 1.0f);
  }
}

__global__ void k_dinv(float* __restrict__ deg, int N) {
  int i = blockIdx.x * blockDim.x + threadIdx.x;
  if (i < N) deg[i] = rsqrtf(deg[i]);  // deg >= 1 always (self loop)
}

__global__ void k_norm(const int* __restrict__ src, const int* __restrict__ dst,
                       const float* __restrict__ dinv, float* __restrict__ nrm, int E) {
  int e = blockIdx.x * blockDim.x + threadIdx.x;
  if (e < E) nrm[e] = dinv[src[e]] * dinv[dst[e]];
}

// ---------------------------------------------------------------------------
// GEMM: T = (RELU ? relu(A) : A) @ W   via V_WMMA_F32_16X16X4_F32
// One wave32 computes one 16x16 tile of T.  N multiple of 16 (100000 = 6250*16).
// Compile-time DIN/DOUT/RELU -> fully unrolled K loop, no flag branches.
// A layout (16x4 f32): lane L holds A[m0 + L%16][k0 + 2*(L/16) + {0,1}]
// B layout (4x16 f32): lane L holds W[k0 + 2*(L/16) + {0,1}][n0 + L%16]
// C/D layout: c[v] = T[m0 + v + 8*(L/16)][n0 + L%16]
// ---------------------------------------------------------------------------
template <int DIN, int DOUT, bool RELU>
__global__ void k_gemm_wmma(const float* __restrict__ A, const float* __restrict__ W,
                            float* __restrict__ T, int mtiles) {
  constexpr int NTILES = DOUT / 16;
  int gid  = blockIdx.x * blockDim.x + threadIdx.x;
  int wave = gid >> 5;
  int lane = gid & 31;
  if (wave >= mtiles * NTILES) return;  // wave-uniform: EXEC stays all-1s

  int m0 = (wave / NTILES) << 4;
  int n0 = (wave % NTILES) << 4;
  int mr = lane & 15;         // row-in-tile for A, col-in-tile for B/D
  int kk = (lane >> 4) << 1;  // K sub-offset: 0 for lanes 0-15, 2 for 16-31

  const float* Arow = A + (size_t)(m0 + mr) * DIN;  // 8B-aligned (DIN % 4 == 0)
  const float* Wcol = W + n0 + mr;

  v8f c = {0.f, 0.f, 0.f, 0.f, 0.f, 0.f, 0.f, 0.f};

#pragma unroll
  for (int k0 = 0; k0 < DIN; k0 += 4) {
    int ka = k0 + kk;
    v2f a = *(const v2f*)(Arow + ka);  // contiguous pair -> global_load_b64
    if (RELU) {
      a.x = fmaxf(a.x, 0.0f);
      a.y = fmaxf(a.y, 0.0f);
    }
    v2f b;
    b.x = Wcol[(size_t)ka * DOUT];
    b.y = Wcol[(size_t)(ka + 1) * DOUT];
    c = __builtin_amdgcn_wmma_f32_16x16x4_f32(
        /*neg_a=*/false, a, /*neg_b=*/false, b,
        /*c_mod=*/(short)0, c, /*reuse_a=*/false, /*reuse_b=*/false);
  }

  float* Trow = T + (size_t)(m0 + ((lane >> 4) << 3)) * DOUT + n0 + mr;
#pragma unroll
  for (int v = 0; v < 8; ++v) {
    Trow[(size_t)v * DOUT] = c[v];
  }
}

// ---------------------------------------------------------------------------
// out[i][f] = b[f] + T[i][f] * dinv[i]^2      (bias + self-loop message)
// ---------------------------------------------------------------------------
template <int DOUT>
__global__ void k_init_out(const float* __restrict__ T, const float* __restrict__ b,
                           const float* __restrict__ dinv, float* __restrict__ out,
                           int N) {
  int i = blockIdx.x * blockDim.x + threadIdx.x;
  if (i < N * DOUT) {
    int node = i / DOUT;        // constant -> shift
    int f    = i & (DOUT - 1);  // DOUT is a power of two
    float dv = dinv[node];
    out[i] = b[f] + T[i] * dv * dv;
  }
}

// ---------------------------------------------------------------------------
// out[dst[e]][f] += T[src[e]][f] * norm[e]
// Thread (e, f): feature = lane -> coalesced gather + native f32 atomics.
// For DOUT==32 a whole wave handles one edge (src/dst/norm wave-uniform).
// ---------------------------------------------------------------------------
template <int DOUT>
__global__ void k_scatter(const int* __restrict__ src, const int* __restrict__ dst,
                          const float* __restrict__ nrm, const float* __restrict__ T,
                          float* __restrict__ out, int E) {
  long long i = (long long)blockIdx.x * blockDim.x + threadIdx.x;
  if (i < (long long)E * DOUT) {
    int e = (int)(i / DOUT);        // constant -> shift
    int f = (int)(i & (DOUT - 1));  // power of two
    float v = T[(size_t)src[e] * DOUT + f] * nrm[e];
    unsafeAtomicAdd(&out[(size_t)dst[e] * DOUT + f], v);
  }
}

// ---------------------------------------------------------------------------
// Host-side orchestration
// ---------------------------------------------------------------------------
static inline int cdiv(long long a, int b) { return (int)((a + b - 1) / b); }

extern "C" void kernel_launch(void* const* d_in, const int* in_sizes, int n_in,
                              void* d_out, int out_size, void* d_ws, size_t ws_size,
                              hipStream_t stream) {
  const float*     x  = (const float*)d_in[0];
  const long long* ei = (const long long*)d_in[1];
  const float* W[6];
  const float* B[6];
  for (int i = 0; i < 6; ++i) {
    W[i] = (const float*)d_in[2 + 2 * i];
    B[i] = (const float*)d_in[3 + 2 * i];
  }
  const int N = in_sizes[0] / 128;  // 100000
  const int E = in_sizes[1] / 2;    // 3200000

  // Workspace carve-up (256B aligned slices)
  char* p = (char*)d_ws;
  auto carve = [&](size_t bytes) -> char* {
    char* r = p;
    p += (bytes + 255) & ~(size_t)255;
    return r;
  };
  int*   src32 = (int*)  carve(sizeof(int)   * (size_t)E);
  int*   dst32 = (int*)  carve(sizeof(int)   * (size_t)E);
  float* nrm   = (float*)carve(sizeof(float) * (size_t)E);
  float* dinv  = (float*)carve(sizeof(float) * (size_t)N);       // deg -> dinv in place
  float* bufA  = (float*)carve(sizeof(float) * (size_t)N * 32);  // GEMM output T
  float* bufB  = (float*)carve(sizeof(float) * (size_t)N * 32);  // h ping
  float* bufC  = (float*)carve(sizeof(float) * (size_t)N * 32);  // h pong
  (void)ws_size;

  const int TB = 256;
  const int mtiles = N / 16;  // exact: 100000 = 6250 * 16

  // ---- normalization preprocessing (once) ----
  k_init_deg<<<cdiv(N, TB), TB, 0, stream>>>(dinv, N);
  k_edge_prep<<<cdiv(E, TB), TB, 0, stream>>>(ei, src32, dst32, dinv, E);
  k_dinv<<<cdiv(N, TB), TB, 0, stream>>>(dinv, N);
  k_norm<<<cdiv(E, TB), TB, 0, stream>>>(src32, dst32, dinv, nrm, E);

  auto gemm_grid = [&](int dout) { return cdiv((long long)mtiles * (dout / 16) * 32, TB); };

  // ---- layer 0: din=128, dout=32, no input relu ----
  k_gemm_wmma<128, 32, false><<<gemm_grid(32), TB, 0, stream>>>(x, W[0], bufA, mtiles);
  k_init_out<32><<<cdiv((long long)N * 32, TB), TB, 0, stream>>>(bufA, B[0], dinv, bufB, N);
  k_scatter<32><<<cdiv((long long)E * 32, TB), TB, 0, stream>>>(src32, dst32, nrm, bufA, bufB, E);

  // ---- layers 1..4: din=32, dout=32, relu on input ----
  const float* hin = bufB;
  float* hping[2] = {bufC, bufB};  // layer l (1-based here) alternates outputs
  for (int l = 1; l <= 4; ++l) {
    float* out = hping[(l - 1) & 1];
    k_gemm_wmma<32, 32, true><<<gemm_grid(32), TB, 0, stream>>>(hin, W[l], bufA, mtiles);
    k_init_out<32><<<cdiv((long long)N * 32, TB), TB, 0, stream>>>(bufA, B[l], dinv, out, N);
    k_scatter<32><<<cdiv((long long)E * 32, TB), TB, 0, stream>>>(src32, dst32, nrm, bufA, out, E);
    hin = out;
  }

  // ---- layer 5: din=32, dout=16, relu on input, write d_out ----
  float* outF = (float*)d_out;
  k_gemm_wmma<32, 16, true><<<gemm_grid(16), TB, 0, stream>>>(hin, W[5], bufA, mtiles);
  k_init_out<16><<<cdiv((long long)N * 16, TB), TB, 0, stream>>>(bufA, B[5], dinv, outF, N);
  k_scatter<16><<<cdiv((long long)E * 16, TB), TB, 0, stream>>>(src32, dst32, nrm, bufA, outF, E);
}